// LieNet_36799279792877
// MI455X (gfx1250) — compile-verified
//
#include <hip/hip_runtime.h>

typedef float v2f __attribute__((ext_vector_type(2)));
typedef float v8f __attribute__((ext_vector_type(8)));

#define NN 256
#define MM 342
#define MH 171
#define FF 64

__device__ __forceinline__ float angle_of(float tr) {
    const float EPS  = 1e-12f;
    const float PI_F = 3.14159265358979323846f;
    bool maskpi   = (fabsf(tr + 1.0f) > EPS) && (fabsf(tr - 3.0f) <= EPS);
    bool maskacos = fabsf(tr + 1.0f) <= EPS;
    float clipped = fminf(fmaxf(tr, -1.0f + 1e-7f), 1.0f - 1e-7f);
    float acosv   = acosf(clipped);
    return tr * (maskpi ? 1.0f : 0.0f) * PI_F + (maskacos ? acosv : 0.0f);
}

// Stage 1+2: rotmap(w1) fused with pool_space (pairs over M).
// One wave handles (n, m-pair, f-group of 16): 6 WMMAs (2 m x 3 k).
__global__ __launch_bounds__(256)
void k_rot1_poolspace(const float* __restrict__ x, const float* __restrict__ w1,
                      float* __restrict__ h1) {
    const int wave = blockIdx.x * (blockDim.x >> 5) + (threadIdx.x >> 5);
    const int lane = threadIdx.x & 31;
    const int fg = wave & 3;             // 4 f-groups of 16
    const int mp = (wave >> 2) % MH;     // 171 m-pairs
    const int n  = wave / (4 * MH);
    const int f0 = fg << 4;
    const int c  = lane & 15;            // column (f offset) / A row (i)
    const int jp = lane >> 4;            // 0: K=j in {0,1}; 1: K=j in {2,3(zero)}
    const int m0 = 2 * mp;
    const int m1 = m0 + 1;

    // A = w1[i, j, m] padded to 16x4 (rows >=3 and j==3 are zero)
    v2f a0 = {0.0f, 0.0f}, a1 = {0.0f, 0.0f};
    if (c < 3) {
        a0[0] = w1[(c * 3 + 2 * jp) * MM + m0];
        a1[0] = w1[(c * 3 + 2 * jp) * MM + m1];
        if (jp == 0) {
            a0[1] = w1[(c * 3 + 1) * MM + m0];
            a1[1] = w1[(c * 3 + 1) * MM + m1];
        }
    }

    v8f d0[3], d1[3];
#pragma unroll
    for (int k = 0; k < 3; ++k) {
        // B = x[n, j, k, m, f0+c] as 4x16 (row j==3 zero)
        v2f b0, b1;
        const long base0 = ((long)((n * 3 + 2 * jp) * 3 + k) * MM + m0) * FF + f0 + c;
        b0[0] = x[base0];
        b1[0] = x[base0 + FF];           // m1 = m0+1 -> +FF floats
        if (jp == 0) {
            const long j1 = ((long)((n * 3 + 1) * 3 + k) * MM + m0) * FF + f0 + c;
            b0[1] = x[j1];
            b1[1] = x[j1 + FF];
        } else {
            b0[1] = 0.0f;
            b1[1] = 0.0f;
        }
        v8f z = {};
        d0[k] = __builtin_amdgcn_wmma_f32_16x16x4_f32(false, a0, false, b0,
                                                      (short)0, z, false, false);
        d1[k] = __builtin_amdgcn_wmma_f32_16x16x4_f32(false, a1, false, b1,
                                                      (short)0, z, false, false);
    }

    // trace lives in-lane: r[i,i] = d[k=i][vgpr i] (lanes 0..15)
    float tr0 = d0[0][0] + d0[1][1] + d0[2][2];
    float tr1 = d1[0][0] + d1[1][1] + d1[2][2];
    float ang0 = angle_of(tr0);
    float ang1 = angle_of(tr1);
    bool sel1 = ang1 > ang0;             // argmax, ties -> first (m0)

    if (lane < 16) {
#pragma unroll
        for (int i = 0; i < 3; ++i)
#pragma unroll
            for (int k = 0; k < 3; ++k) {
                float v = sel1 ? d1[k][i] : d0[k][i];
                h1[((long)((n * 3 + i) * 3 + k) * MH + mp) * FF + f0 + lane] = v;
            }
    }
}

// Stage 3+4: rotmap(w2) fused with pool_time (groups of 4 over F=64 -> 16).
__global__ __launch_bounds__(256)
void k_rot2_pooltime(const float* __restrict__ h1, const float* __restrict__ w2,
                     float* __restrict__ h2) {
    const int wave = blockIdx.x * (blockDim.x >> 5) + (threadIdx.x >> 5);
    const int lane = threadIdx.x & 31;
    const int fg = wave & 3;
    const int m  = (wave >> 2) % MH;
    const int n  = wave / (4 * MH);
    const int f0 = fg << 4;
    const int c  = lane & 15;
    const int jp = lane >> 4;

    v2f a = {0.0f, 0.0f};
    if (c < 3) {
        a[0] = w2[(c * 3 + 2 * jp) * MH + m];
        if (jp == 0) a[1] = w2[(c * 3 + 1) * MH + m];
    }

    v8f d[3];
#pragma unroll
    for (int k = 0; k < 3; ++k) {
        v2f b;
        b[0] = h1[((long)((n * 3 + 2 * jp) * 3 + k) * MH + m) * FF + f0 + c];
        b[1] = (jp == 0)
                 ? h1[((long)((n * 3 + 1) * 3 + k) * MH + m) * FF + f0 + c]
                 : 0.0f;
        v8f z = {};
        d[k] = __builtin_amdgcn_wmma_f32_16x16x4_f32(false, a, false, b,
                                                     (short)0, z, false, false);
    }

    float tr  = d[0][0] + d[1][1] + d[2][2];
    float ang = angle_of(tr);

    // argmax over each group of 4 consecutive f (lanes), ties -> first
    int base = lane & ~3;
    float g0 = __shfl(ang, base + 0, 32);
    float g1 = __shfl(ang, base + 1, 32);
    float g2 = __shfl(ang, base + 2, 32);
    float g3 = __shfl(ang, base + 3, 32);
    int idx = 0; float best = g0;
    if (g1 > best) { best = g1; idx = 1; }
    if (g2 > best) { best = g2; idx = 2; }
    if (g3 > best) { best = g3; idx = 3; }

    if (lane < 16 && (lane & 3) == idx) {
        const int fo = (f0 >> 2) + (lane >> 2);
#pragma unroll
        for (int i = 0; i < 3; ++i)
#pragma unroll
            for (int k = 0; k < 3; ++k)
                h2[((long)((n * 3 + i) * 3 + k) * MH + m) * 16 + fo] = d[k][i];
    }
}

// Stage 5+6: rotmap(w3) fused with pool_time (F=16 -> 4).
__global__ __launch_bounds__(256)
void k_rot3_pooltime(const float* __restrict__ h2, const float* __restrict__ w3,
                     float* __restrict__ h3) {
    const int wave = blockIdx.x * (blockDim.x >> 5) + (threadIdx.x >> 5);
    const int lane = threadIdx.x & 31;
    const int m = wave % MH;
    const int n = wave / MH;
    const int c  = lane & 15;
    const int jp = lane >> 4;

    v2f a = {0.0f, 0.0f};
    if (c < 3) {
        a[0] = w3[(c * 3 + 2 * jp) * MH + m];
        if (jp == 0) a[1] = w3[(c * 3 + 1) * MH + m];
    }

    v8f d[3];
#pragma unroll
    for (int k = 0; k < 3; ++k) {
        v2f b;
        b[0] = h2[((long)((n * 3 + 2 * jp) * 3 + k) * MH + m) * 16 + c];
        b[1] = (jp == 0)
                 ? h2[((long)((n * 3 + 1) * 3 + k) * MH + m) * 16 + c]
                 : 0.0f;
        v8f z = {};
        d[k] = __builtin_amdgcn_wmma_f32_16x16x4_f32(false, a, false, b,
                                                     (short)0, z, false, false);
    }

    float tr  = d[0][0] + d[1][1] + d[2][2];
    float ang = angle_of(tr);

    int base = lane & ~3;
    float g0 = __shfl(ang, base + 0, 32);
    float g1 = __shfl(ang, base + 1, 32);
    float g2 = __shfl(ang, base + 2, 32);
    float g3 = __shfl(ang, base + 3, 32);
    int idx = 0; float best = g0;
    if (g1 > best) { best = g1; idx = 1; }
    if (g2 > best) { best = g2; idx = 2; }
    if (g3 > best) { best = g3; idx = 3; }

    if (lane < 16 && (lane & 3) == idx) {
        const int fo = lane >> 2;
#pragma unroll
        for (int i = 0; i < 3; ++i)
#pragma unroll
            for (int k = 0; k < 3; ++k)
                h3[((long)((n * 3 + i) * 3 + k) * MH + m) * 4 + fo] = d[k][i];
    }
}

// Stage 7: logmap + relu -> d_out (256, 2736)
__global__ __launch_bounds__(256)
void k_logmap_relu(const float* __restrict__ h3, float* __restrict__ out) {
    const int t = blockIdx.x * blockDim.x + threadIdx.x;
    if (t >= NN * MH * 4) return;
    const int f = t & 3;
    const int m = (t >> 2) % MH;
    const int n = t / (4 * MH);

    float r[3][3];
#pragma unroll
    for (int i = 0; i < 3; ++i)
#pragma unroll
        for (int k = 0; k < 3; ++k)
            r[i][k] = h3[((long)((n * 3 + i) * 3 + k) * MH + m) * 4 + f];

    float tr  = r[0][0] + r[1][1] + r[2][2];
    float ang = angle_of(tr);
    float s   = sinf(ang);
    bool safe = fabsf(s) > 1e-12f;
    float ratio = safe ? (ang / (2.0f * s)) : 0.0f;

    float f0 = ratio * (r[2][1] - r[1][2]);
    float f1 = ratio * (r[2][0] - r[0][2]);
    float f2 = ratio * (r[1][0] - r[0][1]);
    float inv = 1.0f / sqrtf(f0 * f0 + f1 * f1 + f2 * f2 + 1e-24f);

    float y[4] = { f0 * inv, f1 * inv, f2 * inv, ang };
    const long obase = (long)n * (MH * 16) + (long)(m * 4 + f) * 4;
#pragma unroll
    for (int cidx = 0; cidx < 4; ++cidx) {
        float v = y[cidx];
        float pos = fmaxf(v, 0.3f)  * ((v > 0.0f) ? 1.0f : 0.0f);
        float neg = fminf(v, -0.3f) * ((v < 0.0f) ? 1.0f : 0.0f);
        out[obase + cidx] = pos + neg;
    }
}

extern "C" void kernel_launch(void* const* d_in, const int* in_sizes, int n_in,
                              void* d_out, int out_size, void* d_ws, size_t ws_size,
                              hipStream_t stream) {
    const float* x  = (const float*)d_in[0];
    const float* w1 = (const float*)d_in[1];
    const float* w2 = (const float*)d_in[2];
    const float* w3 = (const float*)d_in[3];
    float* out = (float*)d_out;

    char* ws = (char*)d_ws;
    const size_t h1_elems = (size_t)NN * 9 * MH * FF;   // 25,214,976 f32
    const size_t h2_elems = (size_t)NN * 9 * MH * 16;   //  6,303,744 f32
    float* h1 = (float*)ws;
    float* h2 = (float*)(ws + h1_elems * sizeof(float));
    float* h3 = (float*)(ws + (h1_elems + h2_elems) * sizeof(float));

    dim3 blk(256);
    const int wavesA = NN * MH * 4;   // 175,104 waves, 8 waves/block
    k_rot1_poolspace<<<wavesA / 8, blk, 0, stream>>>(x, w1, h1);
    k_rot2_pooltime <<<wavesA / 8, blk, 0, stream>>>(h1, w2, h2);
    const int wavesC = NN * MH;       // 43,776 waves
    k_rot3_pooltime <<<wavesC / 8, blk, 0, stream>>>(h2, w3, h3);
    const int nt = NN * MH * 4;
    k_logmap_relu   <<<(nt + 255) / 256, blk, 0, stream>>>(h3, out);
}